// PGnet_64355789964075
// MI455X (gfx1250) — compile-verified
//
#include <hip/hip_runtime.h>

typedef float v2f __attribute__((ext_vector_type(2)));
typedef float v4f __attribute__((ext_vector_type(4)));
typedef float v8f __attribute__((ext_vector_type(8)));

#define NIN   41
#define NL2   4
#define NRC   8
#define BTOT  32768
#define WPB   8                 // waves (= batch elements) per block
#define THREADS (WPB * 32)
#define WELEM (NL2 * NIN * NRC) // 1312 floats per element per weight tensor
#define KSTEPS 22               // K = 88 in steps of 4 for wmma f32 16x16x4

// Wave-synchronous LDS fence: LDS pipe is in-order per wave on CDNA5, so a
// hardware DScnt drain + compiler memory barrier makes cross-lane LDS
// communication within one wave safe without a workgroup barrier.
__device__ __forceinline__ void wave_sync_lds() {
  __builtin_amdgcn_wave_barrier();
  asm volatile("s_wait_dscnt 0" ::: "memory");
  __builtin_amdgcn_wave_barrier();
}

__device__ __forceinline__ float clip_f(float x, float lo, float hi) {
  return fminf(fmaxf(x, lo), hi);
}

__global__ __launch_bounds__(THREADS) void pgnet_kernel(
    const float* __restrict__ inp,  const float* __restrict__ fpgw,
    const float* __restrict__ fpgb, const float* __restrict__ rpgw,
    const float* __restrict__ rpgb, const float* __restrict__ pw,
    const float* __restrict__ pbias, float* __restrict__ out)
{
  // Block-shared: precombined pgctrl weights, laid out as WB[k][n] (K=88 pad, N=16 pad)
  __shared__ float sWB[88 * 16];
  __shared__ float sPB[NRC];
  // Per-wave scratch
  __shared__ float sX [WPB][96];   // x = [inp(41), pad(3), l1i(41), pad..] (zero padded)
  __shared__ float sRC[WPB][8];
  __shared__ float sCF[WPB][32];   // coef[o*8+r] = l2[o]*rc[r]

  const int tid = threadIdx.x;

  // ---- build shared pgctrl operand: WB[k][n] = Wsum[n][k] (k<41), -Wr[n][k-44] (44<=k<85)
  for (int e = tid; e < 88 * 16; e += THREADS) {
    const int k = e >> 4, n = e & 15;
    float v = 0.0f;
    if (n < NRC) {
      if (k < NIN)                  v =  pw[n * 86 + k] + pw[n * 86 + 45 + k];
      else if (k >= 44 && k < 85)   v = -pw[n * 86 + 45 + (k - 44)];
    }
    sWB[e] = v;
  }
  if (tid < NRC) sPB[tid] = pbias[tid];
  __syncthreads();

  const int lane = tid & 31;
  const int wv   = __builtin_amdgcn_readfirstlane(tid >> 5);
  const int b    = (int)blockIdx.x * WPB + wv;

  float* xw = &sX[wv][0];

  // ---- init x vector: [inp | zeros]
  xw[lane]      = inp[(size_t)b * NIN + lane];
  xw[32 + lane] = (lane < 9) ? inp[(size_t)b * NIN + 32 + lane] : 0.0f;
  xw[64 + lane] = 0.0f;

  const int hi = lane >> 4;    // K-half select for WMMA A/B operands
  const int nn = lane & 15;    // N column for B operand

  // ---- hoist the block-constant B operands into VGPRs (used twice per element)
  v2f Bpre[KSTEPS];
#pragma unroll
  for (int t = 0; t < KSTEPS; ++t) {
    const int k0 = 4 * t + 2 * hi;
    Bpre[t][0] = sWB[k0 * 16 + nn];
    Bpre[t][1] = sWB[(k0 + 1) * 16 + nn];
  }

  // ---- cache fpg weights in VGPRs: lane <-> (o = lane>>3, r = lane&7)
  const int og = lane >> 3, rg = lane & 7;
  const float* fwp = fpgw + (size_t)b * WELEM + og * (NIN * NRC) + rg;
  float fw[NIN];
#pragma unroll
  for (int i = 0; i < NIN; ++i) fw[i] = fwp[i * NRC];

  const float fb  = fpgb[(size_t)b * NL2 + og];
  const float rb1 = rpgb[(size_t)b * NIN + lane];
  const float rb2 = (lane < 9) ? rpgb[(size_t)b * NIN + 32 + lane] : 0.0f;

  // rpg bases: inner 32 floats of rpg_w[b][i][o][r] are exactly the (o,r) pairs
  const v4f* rp1 = (const v4f*)(rpgw + (size_t)b * WELEM + (size_t)lane * 32);
  const v4f* rp2 = (const v4f*)(rpgw + (size_t)b * WELEM + (size_t)(32 + lane) * 32);

  float rcOut = 0.0f, l2Out = 0.0f, l1a = 0.0f, l1b = 0.0f;

  for (int it = 0; it < 2; ++it) {
    wave_sync_lds();

    // ---- pgctrl matvec via f32 WMMA chain: D = sum_t A(16x4) x B(4x16)
    // Batch the A broadcast reads so the DS pipe streams them, then run the
    // 22 WMMAs back-to-back with only the C-accumulate dependency.
    v2f Apre[KSTEPS];
#pragma unroll
    for (int t = 0; t < KSTEPS; ++t) {
      const int k0 = 4 * t + 2 * hi;
      Apre[t][0] = xw[k0];
      Apre[t][1] = xw[k0 + 1];
    }
    v8f c = {};
#pragma unroll
    for (int t = 0; t < KSTEPS; ++t) {
      c = __builtin_amdgcn_wmma_f32_16x16x4_f32(false, Apre[t], false, Bpre[t],
                                                (short)0, c, false, false);
    }

    // ---- rc post-processing on lanes 0..7 (n = lane)
    float rv = c[0] + sPB[rg];
    rv = (rv >= 0.0f) ? rv : 0.2f * rv;          // LeakyReLU(0.2)
    rv = clip_f(0.4f * rv, -0.2f, 1.0f);         // 0.4*rc + 0.6*rcold(=0), clip
#pragma unroll
    for (int s = 0; s < 3; ++s) {                // tridiagonal blur x3
      float vm = __shfl(rv, lane - 1, 32);
      float vp = __shfl(rv, lane + 1, 32);
      vm = (lane >= 1 && lane < 8) ? vm : 0.0f;
      vp = (lane < 7)              ? vp : 0.0f;
      rv = 0.8f * rv + 0.1f * (vm + vp);
    }
    if (lane < 8) sRC[wv][lane] = rv;
    rcOut = rv;
    wave_sync_lds();
    const float rcr = sRC[wv][rg];

    // ---- fpg passgate: l2[o] = (sum_{i,r} w*inp[i]*rc[r] + b)*0.1
    float acc = 0.0f;
#pragma unroll
    for (int i = 0; i < NIN; ++i) acc = fmaf(fw[i], xw[i], acc);
    float tsum = acc * rcr;
    tsum += __shfl_xor(tsum, 1, 32);
    tsum += __shfl_xor(tsum, 2, 32);
    tsum += __shfl_xor(tsum, 4, 32);             // 8-lane r-group sum (all lanes hold it)
    float l2v = (tsum + fb) * 0.1f;              // + 0.9*l2old(=0)
    l2v = clip_f(l2v, -0.2f, 1.0f);
#pragma unroll
    for (int s = 0; s < 2; ++s) {                // blur over 4 l2 groups x2
      float vm = __shfl(l2v, lane - 8, 32);
      float vp = __shfl(l2v, (lane + 8) & 31, 32);
      vm = (og > 0) ? vm : 0.0f;
      vp = (og < 3) ? vp : 0.0f;
      l2v = 0.8f * l2v + 0.1f * (vm + vp);
    }
    l2Out = l2v;
    sCF[wv][lane] = rcr * l2v;                   // coef[o*8+r]
    wave_sync_lds();

    // ---- rpg passgate: l1i[i] = clip(sum_j rw[i][j]*coef[j] + rb, -0.2, 1)
    const v4f* cf4 = (const v4f*)&sCF[wv][0];
    float a1 = 0.0f;
#pragma unroll
    for (int q = 0; q < 8; ++q) {
      v4f w = rp1[q];
      v4f cv = cf4[q];
      a1 = fmaf(w[0], cv[0], fmaf(w[1], cv[1], fmaf(w[2], cv[2], fmaf(w[3], cv[3], a1))));
    }
    l1a = clip_f(a1 + rb1, -0.2f, 1.0f);
    xw[44 + lane] = l1a;                         // feeds next iteration's rcin (-Wr folded into WB)
    if (lane < 9) {                              // remaining outputs i = 32..40
      float a2 = 0.0f;
#pragma unroll
      for (int q = 0; q < 8; ++q) {
        v4f w = rp2[q];
        v4f cv = cf4[q];
        a2 = fmaf(w[0], cv[0], fmaf(w[1], cv[1], fmaf(w[2], cv[2], fmaf(w[3], cv[3], a2))));
      }
      l1b = clip_f(a2 + rb2, -0.2f, 1.0f);
      xw[44 + 32 + lane] = l1b;
    }
  }

  // ---- outputs: [l1i (B*41) | l2 (B*4) | rc (B*8)]
  float* outL1 = out;
  float* outL2 = out + (size_t)BTOT * NIN;
  float* outRC = out + (size_t)BTOT * (NIN + NL2);
  outL1[(size_t)b * NIN + lane] = l1a;
  if (lane < 9) outL1[(size_t)b * NIN + 32 + lane] = l1b;
  if (rg == 0)  outL2[(size_t)b * NL2 + og] = l2Out;
  if (lane < 8) outRC[(size_t)b * NRC + lane] = rcOut;
}

extern "C" void kernel_launch(void* const* d_in, const int* in_sizes, int n_in,
                              void* d_out, int out_size, void* d_ws, size_t ws_size,
                              hipStream_t stream) {
  (void)in_sizes; (void)n_in; (void)out_size; (void)d_ws; (void)ws_size;
  const float* inp  = (const float*)d_in[0];
  const float* fpgw = (const float*)d_in[1];
  const float* fpgb = (const float*)d_in[2];
  const float* rpgw = (const float*)d_in[3];
  const float* rpgb = (const float*)d_in[4];
  const float* pw   = (const float*)d_in[5];
  const float* pb   = (const float*)d_in[6];
  float* out = (float*)d_out;

  dim3 grid(BTOT / WPB);
  pgnet_kernel<<<grid, THREADS, 0, stream>>>(inp, fpgw, fpgb, rpgw, rpgb, pw, pb, out);
}